// DynamicTGNEncoder_58334245814660
// MI455X (gfx1250) — compile-verified
//
#include <hip/hip_runtime.h>
#include <math.h>

// Problem constants (from reference)
constexpr int N_NODES   = 50000;
constexpr int N_EDGES   = 800000;
constexpr int IN_DIM    = 128;
constexpr int D         = 32;            // MEM_DIM
constexpr int ED_STRIDE = 3 + D;         // 35 floats per edge row
constexpr unsigned NEG_INF_ENC = 0x007FFFFFu;  // order-preserving encoding of -inf

typedef float v2f __attribute__((ext_vector_type(2)));
typedef float v8f __attribute__((ext_vector_type(8)));

// ---- order-preserving float <-> uint mapping for atomicMax on floats ----
__device__ __forceinline__ unsigned encf(float f) {
  unsigned u = __float_as_uint(f);
  return (u & 0x80000000u) ? ~u : (u | 0x80000000u);
}
__device__ __forceinline__ float decf(unsigned u) {
  return (u & 0x80000000u) ? __uint_as_float(u & 0x7FFFFFFFu) : __uint_as_float(~u);
}

// ---------------- init workspace state ----------------
__global__ void init_kernel(int* __restrict__ last, unsigned* __restrict__ menc,
                            float* __restrict__ denom, float* __restrict__ aggU,
                            float* __restrict__ zerorow) {
  int i = blockIdx.x * blockDim.x + threadIdx.x;
  if (i < IN_DIM) zerorow[i] = 0.0f;
  if (i < N_NODES) last[i] = -1;
  if (i < 2 * N_NODES) { menc[i] = NEG_INF_ENC; denom[i] = 0.0f; }
  if (i < 64 * N_NODES) aggU[i] = 0.0f;
}

// ---------------- cvec[o] = b_ih[o] + sum_d cos(tw+tb)*W_ih[o, 64+d] ----------------
__global__ void cvec_kernel(const float* __restrict__ time_w, const float* __restrict__ time_b,
                            const float* __restrict__ W_ih, const float* __restrict__ b_ih,
                            float* __restrict__ cvec) {
  int o = threadIdx.x;
  if (o >= 96) return;
  float acc = b_ih[o];
  for (int d2 = 0; d2 < D; ++d2)
    acc = fmaf(cosf(time_w[d2] + time_b[d2]), W_ih[(size_t)o * 96 + 64 + d2], acc);
  cvec[o] = acc;
}

// ---------------- stable compaction of valid nodes (single-block scan) ----------------
__global__ void scan_kernel(const unsigned char* __restrict__ nmask,
                            int* __restrict__ validlist, int* __restrict__ nvp) {
  __shared__ int tmp[1024];
  __shared__ int carry;
  const int t = threadIdx.x;
  if (t == 0) carry = 0;
  __syncthreads();
  for (int base = 0; base < N_NODES; base += 1024) {
    int i = base + t;
    int v = (i < N_NODES && nmask[i]) ? 1 : 0;
    int c = carry;
    __syncthreads();
    tmp[t] = v;
    __syncthreads();
    for (int off = 1; off < 1024; off <<= 1) {
      int y = (t >= off) ? tmp[t - off] : 0;
      __syncthreads();
      tmp[t] += y;
      __syncthreads();
    }
    int incl = tmp[t];
    if (v) validlist[c + incl - 1] = i;
    __syncthreads();
    if (t == 1023) carry = c + incl;
    __syncthreads();
  }
  if (t == 0) *nvp = carry;
}

// ---------------- last[n] = max position of valid edge (src pos=e, dst pos=E+e) ----------------
__global__ void last_kernel(const float* __restrict__ ed, const unsigned char* __restrict__ emask,
                            int* __restrict__ last) {
  int e = blockIdx.x * blockDim.x + threadIdx.x;
  if (e >= N_EDGES) return;
  if (!emask[e]) return;
  const float* row = ed + (size_t)e * ED_STRIDE;
  int s = (int)row[0]; s = min(max(s, 0), N_NODES - 1);
  int d = (int)row[1]; d = min(max(d, 0), N_NODES - 1);
  atomicMax(last + s, e);
  atomicMax(last + d, N_EDGES + e);
}

// ---------------- GRU(h=0) memory: one thread per (node, d) ----------------
__global__ void gru_mem_kernel(const float* __restrict__ ed, const int* __restrict__ last,
                               const float* __restrict__ W_ih, const float* __restrict__ b_hh,
                               const float* __restrict__ cvec, float* __restrict__ mem) {
  int t = blockIdx.x * blockDim.x + threadIdx.x;
  if (t >= N_NODES * D) return;
  int n = t >> 5, d = t & 31;
  int ls = last[n];
  if (ls < 0) { mem[t] = 0.0f; return; }
  int li = ls % N_EDGES;
  const float* raw = ed + (size_t)li * ED_STRIDE + 3;
  float gr = cvec[d], gz = cvec[32 + d], gn = cvec[64 + d];
  const float* wr = W_ih + (size_t)d * 96 + 32;
  const float* wz = W_ih + (size_t)(32 + d) * 96 + 32;
  const float* wn = W_ih + (size_t)(64 + d) * 96 + 32;
#pragma unroll 8
  for (int dd = 0; dd < D; ++dd) {
    float x = raw[dd];
    gr = fmaf(x, wr[dd], gr);
    gz = fmaf(x, wz[dd], gz);
    gn = fmaf(x, wn[dd], gn);
  }
  float r  = 1.0f / (1.0f + __expf(-(gr + b_hh[d])));
  float z  = 1.0f / (1.0f + __expf(-(gz + b_hh[32 + d])));
  float nn = tanhf(gn + r * b_hh[64 + d]);
  mem[t] = (1.0f - z) * nn;
}

// ---------------- projection GEMM via V_WMMA_F32_16X16X4_F32 ----------------
// One wave per 16x16 output tile; K=128 fully unrolled into 32 chained WMMAs.
// NCOL is compile-time so all A/B/D addresses fold to immediate offsets.
// Invalid rows (>= num_valid) read from a zero-filled dummy row: no exec divergence.
template <int NCOL>
__global__ void __launch_bounds__(32)
proj_gemm_kernel(const float* __restrict__ nf, const int* __restrict__ validlist,
                 const int* __restrict__ nvp, const float* __restrict__ zerorow,
                 const float* __restrict__ W, float* __restrict__ Out) {
  const int lane   = threadIdx.x;      // 0..31 (wave32)
  const int tileM  = blockIdx.x;       // 50000/16 = 3125 row tiles
  const int tileN  = blockIdx.y;       // NCOL/16 col tiles
  const int nlocal = lane & 15;
  const int khalf  = lane >> 4;        // lanes 16..31 carry K+2,K+3 per ISA A/B layout
  const int nv = *nvp;

  const int rowm = tileM * 16 + nlocal;            // A-matrix row this lane feeds
  const float* rp = (rowm < nv) ? (nf + (size_t)validlist[rowm] * IN_DIM + 2 * khalf)
                                : (zerorow + 2 * khalf);
  const float* bp = W + (size_t)(2 * khalf) * NCOL + tileN * 16 + nlocal;

  v8f c = {};
#pragma unroll
  for (int kk = 0; kk < IN_DIM; kk += 4) {
    v2f a = *(const v2f*)(rp + kk);                // contiguous pair -> global_load_b64
    v2f b;
    b.x = bp[(size_t)kk * NCOL];                   // immediate-offset b32 loads
    b.y = bp[(size_t)(kk + 1) * NCOL];
    c = __builtin_amdgcn_wmma_f32_16x16x4_f32(false, a, false, b, (short)0, c, false, false);
  }
  // C/D layout: lanes 0-15 hold M = r, lanes 16-31 hold M = 8 + r (N = lane & 15)
  float* op = Out + (size_t)(tileM * 16 + 8 * khalf) * NCOL + tileN * 16 + nlocal;
#pragma unroll
  for (int r = 0; r < 8; ++r)
    op[(size_t)r * NCOL] = c[r];
}

// ---------------- pass A: s = scale * <q[dst], k[src]>, segment max via encoded atomicMax ----------------
__global__ void edge_score_kernel(const float* __restrict__ ed, const unsigned char* __restrict__ emask,
                                  const float* __restrict__ Qb, const float* __restrict__ Kb,
                                  float* __restrict__ sbuf, int2* __restrict__ idxbuf,
                                  unsigned* __restrict__ menc) {
  int e = blockIdx.x * blockDim.x + threadIdx.x;
  if (e >= N_EDGES) return;
  const float* row = ed + (size_t)e * ED_STRIDE;
  int s = (int)row[0]; s = min(max(s, 0), N_NODES - 1);
  int d = (int)row[1]; d = min(max(d, 0), N_NODES - 1);
  idxbuf[e] = make_int2(s, d);
  if (!emask[e]) { sbuf[2 * e] = -INFINITY; sbuf[2 * e + 1] = -INFINITY; return; }
  const float4* qd = (const float4*)(Qb + (size_t)d * 64);
  const float4* ks = (const float4*)(Kb + (size_t)s * 64);
  float a0 = 0.0f, a1 = 0.0f;
#pragma unroll
  for (int i = 0; i < 8; ++i) {
    float4 x = qd[i], y = ks[i];
    a0 += x.x * y.x + x.y * y.y + x.z * y.z + x.w * y.w;
  }
#pragma unroll
  for (int i = 8; i < 16; ++i) {
    float4 x = qd[i], y = ks[i];
    a1 += x.x * y.x + x.y * y.y + x.z * y.z + x.w * y.w;
  }
  const float scale = 0.17677669529663687f;  // 1/sqrt(32)
  a0 *= scale; a1 *= scale;
  sbuf[2 * e] = a0; sbuf[2 * e + 1] = a1;
  atomicMax(menc + 2 * d, encf(a0));
  atomicMax(menc + 2 * d + 1, encf(a1));
}

// ---------------- pass B: aggU[dst] += p*v[src], denom[dst] += p ----------------
__global__ void edge_agg_kernel(const unsigned char* __restrict__ emask, const float* __restrict__ sbuf,
                                const int2* __restrict__ idxbuf, const unsigned* __restrict__ menc,
                                const float* __restrict__ Vb, float* __restrict__ denom,
                                float* __restrict__ aggU) {
  int e = blockIdx.x * blockDim.x + threadIdx.x;
  if (e >= N_EDGES) return;
  if (!emask[e]) return;
  int2 sd = idxbuf[e];
  int s = sd.x, d = sd.y;
  float m0 = decf(menc[2 * d]);     m0 = isfinite(m0) ? m0 : 0.0f;
  float m1 = decf(menc[2 * d + 1]); m1 = isfinite(m1) ? m1 : 0.0f;
  float p0 = __expf(sbuf[2 * e] - m0);
  float p1 = __expf(sbuf[2 * e + 1] - m1);
  atomicAdd(denom + 2 * d, p0);
  atomicAdd(denom + 2 * d + 1, p1);
  const float* vs = Vb + (size_t)s * 64;
  float* ag = aggU + (size_t)d * 64;
#pragma unroll 8
  for (int i = 0; i < D; ++i) {
    atomicAdd(ag + i,      p0 * vs[i]);
    atomicAdd(ag + 32 + i, p1 * vs[32 + i]);
  }
}

// ---------------- out = mean_h(aggU/denom) + skip + front*memory ----------------
__global__ void out_kernel(const float* __restrict__ aggU, const float* __restrict__ denom,
                           const float* __restrict__ SK, const float* __restrict__ mem,
                           const int* __restrict__ nvp, float* __restrict__ out) {
  int t = blockIdx.x * blockDim.x + threadIdx.x;
  if (t >= N_NODES * D) return;
  int n = t >> 5, d = t & 31;
  float den0 = denom[2 * n] + 1e-16f;
  float den1 = denom[2 * n + 1] + 1e-16f;
  float a = 0.5f * (aggU[(size_t)n * 64 + d] / den0 + aggU[(size_t)n * 64 + 32 + d] / den1);
  float o = a + SK[t];
  if (n < *nvp) o += mem[t];
  out[t] = o;
}

extern "C" void kernel_launch(void* const* d_in, const int* in_sizes, int n_in,
                              void* d_out, int out_size, void* d_ws, size_t ws_size,
                              hipStream_t stream) {
  const float* edge_data       = (const float*)d_in[0];
  const float* node_features   = (const float*)d_in[1];
  const unsigned char* emask   = (const unsigned char*)d_in[2];
  const unsigned char* nmask   = (const unsigned char*)d_in[3];
  const float* time_w          = (const float*)d_in[4];
  const float* time_b          = (const float*)d_in[5];
  const float* W_ih            = (const float*)d_in[6];
  // d_in[7] = W_hh: dead (h == 0 -> gh == b_hh)
  const float* b_ih            = (const float*)d_in[8];
  const float* b_hh            = (const float*)d_in[9];
  const float* Wq              = (const float*)d_in[10];
  const float* Wk              = (const float*)d_in[11];
  const float* Wv              = (const float*)d_in[12];
  const float* Wskip           = (const float*)d_in[13];
  float* out = (float*)d_out;
  (void)in_sizes; (void)n_in; (void)out_size; (void)ws_size;

  char* w = (char*)d_ws;
  size_t off = 0;
  auto take = [&](size_t bytes) { size_t r = off; off = (off + bytes + 255) & ~(size_t)255; return r; };
  int*      last      = (int*)     (w + take(sizeof(int) * N_NODES));
  int*      validlist = (int*)     (w + take(sizeof(int) * N_NODES));
  int*      nvp       = (int*)     (w + take(256));
  float*    cvec      = (float*)   (w + take(sizeof(float) * 96));
  float*    zerorow   = (float*)   (w + take(sizeof(float) * IN_DIM));
  float*    mem       = (float*)   (w + take(sizeof(float) * N_NODES * D));
  float*    Qb        = (float*)   (w + take(sizeof(float) * N_NODES * 64));
  float*    Kb        = (float*)   (w + take(sizeof(float) * N_NODES * 64));
  float*    Vb        = (float*)   (w + take(sizeof(float) * N_NODES * 64));
  float*    SK        = (float*)   (w + take(sizeof(float) * N_NODES * D));
  float*    sbuf      = (float*)   (w + take(sizeof(float) * N_EDGES * 2));
  int2*     idxbuf    = (int2*)    (w + take(sizeof(int2) * N_EDGES));
  unsigned* menc      = (unsigned*)(w + take(sizeof(unsigned) * N_NODES * 2));
  float*    denom     = (float*)   (w + take(sizeof(float) * N_NODES * 2));
  float*    aggU      = (float*)   (w + take(sizeof(float) * N_NODES * 64));

  init_kernel<<<(64 * N_NODES + 255) / 256, 256, 0, stream>>>(last, menc, denom, aggU, zerorow);
  cvec_kernel<<<1, 96, 0, stream>>>(time_w, time_b, W_ih, b_ih, cvec);
  scan_kernel<<<1, 1024, 0, stream>>>(nmask, validlist, nvp);
  last_kernel<<<(N_EDGES + 255) / 256, 256, 0, stream>>>(edge_data, emask, last);
  gru_mem_kernel<<<(N_NODES * D + 255) / 256, 256, 0, stream>>>(edge_data, last, W_ih, b_hh, cvec, mem);

  dim3 g64(N_NODES / 16, 64 / 16);   // 3125 x 4
  dim3 g32(N_NODES / 16, 32 / 16);   // 3125 x 2
  proj_gemm_kernel<64><<<g64, 32, 0, stream>>>(node_features, validlist, nvp, zerorow, Wq, Qb);
  proj_gemm_kernel<64><<<g64, 32, 0, stream>>>(node_features, validlist, nvp, zerorow, Wk, Kb);
  proj_gemm_kernel<64><<<g64, 32, 0, stream>>>(node_features, validlist, nvp, zerorow, Wv, Vb);
  proj_gemm_kernel<32><<<g32, 32, 0, stream>>>(node_features, validlist, nvp, zerorow, Wskip, SK);

  edge_score_kernel<<<(N_EDGES + 255) / 256, 256, 0, stream>>>(edge_data, emask, Qb, Kb,
                                                               sbuf, idxbuf, menc);
  edge_agg_kernel<<<(N_EDGES + 255) / 256, 256, 0, stream>>>(emask, sbuf, idxbuf, menc,
                                                             Vb, denom, aggU);
  out_kernel<<<(N_NODES * D + 255) / 256, 256, 0, stream>>>(aggU, denom, SK, mem, nvp, out);
}